// Conv1D_style_74191265071781
// MI455X (gfx1250) — compile-verified
//
#include <hip/hip_runtime.h>
#include <hip/hip_bf16.h>

typedef __attribute__((ext_vector_type(2))) float v2f;
typedef __attribute__((ext_vector_type(8))) float v8f;

// Problem constants (from reference setup_inputs)
#define BB   8
#define TT   1024
#define NX   1024
#define NF   4096
#define KK   50

#define TILE_M 128
#define TILE_N 128
#define TILE_K 32
#define LDS_STRIDE 36   // 36*m mod 64 -> 16 distinct multiples of 4; float2 frag loads bank-conflict-free
#define EP_STRIDE  68   // epilogue staging stride: 8*68 mod 64 = 32 -> lane-halves hit disjoint bank sets

// ---------------------------------------------------------------------------
// Pre-kernel: tmpL = cluster @ style_L  (B, NX); tmpR = cluster @ style_R (B, NF)
// ---------------------------------------------------------------------------
__global__ __launch_bounds__(256) void style_precompute_kernel(
    const float* __restrict__ cluster,   // (B, KK)
    const float* __restrict__ style_L,   // (KK, NX)
    const float* __restrict__ style_R,   // (KK, NF)
    float* __restrict__ tmpL,            // (B, NX)
    float* __restrict__ tmpR)            // (B, NF)
{
    int i = blockIdx.x * blockDim.x + threadIdx.x;
    const int nL = BB * NX;
    const int nR = BB * NF;
    if (i < nL) {
        int b = i / NX, c = i % NX;
        float s = 0.f;
        #pragma unroll
        for (int k = 0; k < KK; ++k)
            s = fmaf(cluster[b * KK + k], style_L[k * NX + c], s);
        tmpL[i] = s;
    } else if (i < nL + nR) {
        int j = i - nL;
        int b = j / NF, c = j % NF;
        float s = 0.f;
        #pragma unroll
        for (int k = 0; k < KK; ++k)
            s = fmaf(cluster[b * KK + k], style_R[k * NF + c], s);
        tmpR[j] = s;
    }
}

// ---------------------------------------------------------------------------
// Main fused GEMM:
//   out[m, n] = (sum_k x[m,k]*tmpL[b,k]*W[n,k]) * tmpR[b,n] + bias[n]
//   M = B*T = 8192, N = NF = 4096, K = NX = 1024
// 256 threads = 8 wave32; block tile 128x128; wave tile 32x64 (2x4 WMMA accums).
// Software-pipelined: global loads of tile k+1 issue before the WMMAs of tile k.
// Epilogue staged through LDS for full-wave b128 coalesced output stores.
// ---------------------------------------------------------------------------
__global__ __launch_bounds__(256) void conv1d_style_wmma_kernel(
    const float* __restrict__ x,      // (B*T, NX)
    const float* __restrict__ W,      // (NF, NX)
    const float* __restrict__ bias,   // (NF)
    const float* __restrict__ tmpL,   // (B, NX)
    const float* __restrict__ tmpR,   // (B, NF)
    float* __restrict__ out)          // (B*T, NF)
{
    __shared__ union {
        struct {
            float A[TILE_M * LDS_STRIDE];
            float B[TILE_N * LDS_STRIDE];
        } g;
        float ep[8 * 32 * EP_STRIDE];          // 8 waves x 32 rows x 68 floats
    } smem;

    const int m0   = blockIdx.y * TILE_M;
    const int n0   = blockIdx.x * TILE_N;
    const int b    = m0 / TT;          // block's M-rows sit inside one batch (T % TILE_M == 0)
    const int tid  = threadIdx.x;
    const int lane = tid & 31;
    const int wave = tid >> 5;
    const int wm   = (wave & 3) * 32;  // 4 waves along M
    const int wn   = (wave >> 2) * 64; // 2 waves along N
    const int l16  = lane & 15;
    const int koff = (lane >> 4) << 1; // lanes 0-15 -> K{0,1}; lanes 16-31 -> K{2,3}

    // per-thread tile-staging coordinates (128 rows x 32 cols, 4 float4 each)
    const int srow = tid >> 1;              // 0..127   (2 threads per row)
    const int scol = (tid & 1) << 4;        // 0 or 16  (each thread: 4 consecutive float4)

    v8f acc[2][4];
    #pragma unroll
    for (int mt = 0; mt < 2; ++mt)
        #pragma unroll
        for (int nt = 0; nt < 4; ++nt)
            acc[mt][nt] = (v8f)(0.f);

    const float* __restrict__ tmpLb = tmpL + (size_t)b * NX;

    float4 xa[4], wa[4], la[4];

    // prologue: fetch tile kk=0 into registers
    {
        const float* xp = x + (size_t)(m0 + srow) * NX + scol;
        const float* wp = W + (size_t)(n0 + srow) * NX + scol;
        const float* lp = tmpLb + scol;
        #pragma unroll
        for (int i = 0; i < 4; ++i) {
            xa[i] = *(const float4*)(xp + i * 4);
            wa[i] = *(const float4*)(wp + i * 4);
            la[i] = *(const float4*)(lp + i * 4);
        }
    }

    for (int kk = 0; kk < NX; kk += TILE_K) {
        // ---- commit staged registers to LDS (A scaled by tmpL) ----
        {
            float* da = &smem.g.A[srow * LDS_STRIDE + scol];
            float* db = &smem.g.B[srow * LDS_STRIDE + scol];
            #pragma unroll
            for (int i = 0; i < 4; ++i) {
                da[i * 4 + 0] = xa[i].x * la[i].x;
                da[i * 4 + 1] = xa[i].y * la[i].y;
                da[i * 4 + 2] = xa[i].z * la[i].z;
                da[i * 4 + 3] = xa[i].w * la[i].w;
                db[i * 4 + 0] = wa[i].x;
                db[i * 4 + 1] = wa[i].y;
                db[i * 4 + 2] = wa[i].z;
                db[i * 4 + 3] = wa[i].w;
            }
        }
        __syncthreads();

        // ---- issue global loads for next K-tile (overlap with WMMAs below) ----
        if (kk + TILE_K < NX) {                 // uniform branch: EXEC stays all-ones
            const int kn = kk + TILE_K;
            const float* xp = x + (size_t)(m0 + srow) * NX + kn + scol;
            const float* wp = W + (size_t)(n0 + srow) * NX + kn + scol;
            const float* lp = tmpLb + kn + scol;
            #pragma unroll
            for (int i = 0; i < 4; ++i) {
                xa[i] = *(const float4*)(xp + i * 4);
                wa[i] = *(const float4*)(wp + i * 4);
                la[i] = *(const float4*)(lp + i * 4);
            }
        }

        // ---- 8 k-steps of 4; 8 WMMAs per step per wave ----
        #pragma unroll
        for (int k4 = 0; k4 < TILE_K / 4; ++k4) {
            const int kbase = k4 * 4 + koff;
            v2f a[2], bf[4];
            a[0] = *(const v2f*)(&smem.g.A[(wm +      l16) * LDS_STRIDE + kbase]);
            a[1] = *(const v2f*)(&smem.g.A[(wm + 16 + l16) * LDS_STRIDE + kbase]);
            #pragma unroll
            for (int nt = 0; nt < 4; ++nt)
                bf[nt] = *(const v2f*)(&smem.g.B[(wn + nt * 16 + l16) * LDS_STRIDE + kbase]);
            #pragma unroll
            for (int mt = 0; mt < 2; ++mt)
                #pragma unroll
                for (int nt = 0; nt < 4; ++nt)
                    acc[mt][nt] = __builtin_amdgcn_wmma_f32_16x16x4_f32(
                        false, a[mt], false, bf[nt],
                        (short)0, acc[mt][nt], false, false);
        }
        __syncthreads();
    }

    // ---- epilogue: scale by tmpR, add bias, stage in LDS, coalesced b128 stores ----
    const int mhalf = (lane >> 4) << 3;   // C/D layout: VGPR r holds M=r (lanes 0-15) / M=r+8 (lanes 16-31)
    float* ep = &smem.ep[wave * 32 * EP_STRIDE];

    #pragma unroll
    for (int nt = 0; nt < 4; ++nt) {
        const int ncol = n0 + wn + nt * 16 + l16;
        const float rscale = tmpR[(size_t)b * NF + ncol];
        const float bv = bias[ncol];
        #pragma unroll
        for (int mt = 0; mt < 2; ++mt) {
            #pragma unroll
            for (int r = 0; r < 8; ++r) {
                ep[(mt * 16 + mhalf + r) * EP_STRIDE + nt * 16 + l16] =
                    acc[mt][nt][r] * rscale + bv;
            }
        }
    }
    // Same-wave LDS RAW: DS ops are in-order per wave; compiler inserts s_wait_dscnt.
    #pragma unroll
    for (int i = 0; i < 16; ++i) {
        const int row = i * 2 + (lane >> 4);        // lanes 0-15 -> even row, 16-31 -> odd row
        const int col = l16 * 4;                    // full wave covers 2 rows x 256B contiguous
        float4 v = *(const float4*)(&ep[row * EP_STRIDE + col]);
        *(float4*)(out + (size_t)(m0 + wm + row) * NF + (n0 + wn + col)) = v;
    }
}

// ---------------------------------------------------------------------------
extern "C" void kernel_launch(void* const* d_in, const int* in_sizes, int n_in,
                              void* d_out, int out_size, void* d_ws, size_t ws_size,
                              hipStream_t stream) {
    (void)in_sizes; (void)n_in; (void)out_size; (void)ws_size;
    const float* x       = (const float*)d_in[0];   // (8,1024,1024)
    const float* cluster = (const float*)d_in[1];   // (8,50)
    const float* weight  = (const float*)d_in[2];   // (4096,1024)
    const float* bias    = (const float*)d_in[3];   // (4096)
    const float* style_L = (const float*)d_in[4];   // (50,1024)
    const float* style_R = (const float*)d_in[5];   // (50,4096)
    float* out = (float*)d_out;                     // (8,1024,4096)

    float* tmpL = (float*)d_ws;                     // 8*1024 floats
    float* tmpR = tmpL + BB * NX;                   // 8*4096 floats

    {
        int total = BB * NX + BB * NF;              // 40960
        int blocks = (total + 255) / 256;
        style_precompute_kernel<<<blocks, 256, 0, stream>>>(cluster, style_L, style_R, tmpL, tmpR);
    }
    {
        dim3 grid(NF / TILE_N, (BB * TT) / TILE_M); // (32, 64)
        conv1d_style_wmma_kernel<<<grid, 256, 0, stream>>>(x, weight, bias, tmpL, tmpR, out);
    }
}